// HyConv_15573551416001
// MI455X (gfx1250) — compile-verified
//
#include <hip/hip_runtime.h>

// HyConv: out = Dv * H * De * H^T * (x @ theta + bias)
// N=20000 vertices, E=10000 edges, 64 -> 32 channels, all f32.
// Two inherent 800MB passes over H; degree sums fused into the WMMA passes.

typedef __attribute__((ext_vector_type(2))) float v2f;
typedef __attribute__((ext_vector_type(8))) float v8f;

#define NV 20000
#define NE 10000
#define CIN 64
#define COUT 32
#define CH1 1000   // vertex rows per split-K chunk in pass 1 (20 chunks)
#define CH2 500    // edge cols per split-K chunk in pass 2 (20 chunks)

static __device__ __forceinline__ v8f wmma_f32(v2f a, v2f b, v8f c) {
  // D = A(16x4) * B(4x16) + C(16x16), f32, wave32
  return __builtin_amdgcn_wmma_f32_16x16x4_f32(
      /*neg_a=*/false, a, /*neg_b=*/false, b,
      /*c_mod=*/(short)0, c, /*reuse_a=*/false, /*reuse_b=*/false);
}

// ---------------------------------------------------------------------------
// Kernel 1: y = x @ theta + bias   [20000 x 64] * [64 x 32] -> [20000 x 32]
// One wave per 16-row tile; K loop over 64 in steps of 4; two 16-col C tiles.
// ---------------------------------------------------------------------------
__global__ void k_gemm_y(const float* __restrict__ x,
                         const float* __restrict__ theta,
                         const float* __restrict__ bias,
                         float* __restrict__ y) {
  const int l = threadIdx.x;   // 0..31
  const int h = l >> 4;        // half-wave: selects K pair {0,1} vs {2,3}
  const int i = l & 15;        // M (rows of A) / N (cols of B)
  const int m0 = blockIdx.x * 16;

  v8f acc0 = {};
  v8f acc1 = {};
  for (int k0 = 0; k0 < CIN; k0 += 4) {
    // A[m][k] at lane m+16*(k>>1), vgpr k&1 -> contiguous b64 per lane
    v2f a = *(const v2f*)(x + (size_t)(m0 + i) * CIN + k0 + 2 * h);
    const float* tb = theta + (size_t)(k0 + 2 * h) * COUT;
    v2f b0, b1;
    b0.x = tb[i];        b0.y = tb[COUT + i];        // cols 0..15
    b1.x = tb[16 + i];   b1.y = tb[COUT + 16 + i];   // cols 16..31
    acc0 = wmma_f32(a, b0, acc0);
    acc1 = wmma_f32(a, b1, acc1);
  }
  const float bias0 = bias[i];
  const float bias1 = bias[16 + i];
#pragma unroll
  for (int r = 0; r < 8; ++r) {
    const int row = m0 + r + 8 * h;   // C layout: vgpr r -> M = r + 8*half
    y[(size_t)row * COUT + i]      = acc0[r] + bias0;
    y[(size_t)row * COUT + 16 + i] = acc1[r] + bias1;
  }
}

// ---------------------------------------------------------------------------
// Kernel 2 (pass 1): ye_raw[e,c] += sum_v H[v,e]*y[v,c]; colsum[e] += sum_v H[v,e]
// grid = (E/16, 20 split-K chunks); one wave per block. A = H^T tile (16e x 4v).
// ---------------------------------------------------------------------------
__global__ void k_pass1(const float* __restrict__ H,
                        const float* __restrict__ y,
                        float* __restrict__ ye,
                        float* __restrict__ colsum) {
  const int l = threadIdx.x;
  const int h = l >> 4;
  const int i = l & 15;
  const int e0 = blockIdx.x * 16;
  const int vbeg = blockIdx.y * CH1;

  v8f acc0 = {};
  v8f acc1 = {};
  float csum = 0.f;

  const float* Hp = H + (size_t)(vbeg + 2 * h) * NE + e0 + i;  // A: column e0+i
  const float* yp = y + (size_t)(vbeg + 2 * h) * COUT;         // B: rows of y

  for (int it = 0; it < CH1 / 4; ++it) {
    v2f a;
    a.x = Hp[0];              // H[v0+2h+0, e0+i]
    a.y = Hp[NE];             // H[v0+2h+1, e0+i]
    __builtin_prefetch(Hp + (size_t)8 * NE, 0, 0);  // -> global_prefetch_b8
    csum += a.x + a.y;        // fused column-degree partial
    v2f b0, b1;
    b0.x = yp[i];             b0.y = yp[COUT + i];
    b1.x = yp[16 + i];        b1.y = yp[COUT + 16 + i];
    acc0 = wmma_f32(a, b0, acc0);
    acc1 = wmma_f32(a, b1, acc1);
    Hp += (size_t)4 * NE;
    yp += 4 * COUT;
  }

  // lane l and l+16 hold the two K-half partials of column e0+(l&15)
  csum += __shfl_xor(csum, 16, 32);
  if (h == 0) atomicAdd(colsum + e0 + i, csum);

#pragma unroll
  for (int r = 0; r < 8; ++r) {
    const int e = e0 + r + 8 * h;
    atomicAdd(ye + (size_t)e * COUT + i,      acc0[r]);
    atomicAdd(ye + (size_t)e * COUT + 16 + i, acc1[r]);
  }
}

// ---------------------------------------------------------------------------
// Kernel 3 (pass 2): out[v,c] += sum_e H[v,e]*ye[e,c]; rowsum[v] += sum_e H[v,e]
// grid = (N/32, 20 split-K chunks). Each wave owns TWO 16-row M-tiles so the
// L2-resident B fragments (ye) are shared: per 4-K step, 2x b64 A + 4x b32 B
// + 4 WMMA -> 512B of unique HBM traffic per ~15 issue slots.
// ---------------------------------------------------------------------------
__global__ void k_pass2(const float* __restrict__ H,
                        const float* __restrict__ ye,
                        float* __restrict__ out,
                        float* __restrict__ rowsum) {
  const int l = threadIdx.x;
  const int h = l >> 4;
  const int i = l & 15;
  const int v0 = blockIdx.x * 32;          // two 16-row tiles: v0.., v0+16..
  const int ebeg = blockIdx.y * CH2;

  v8f acc00 = {};  // tile0, cols 0..15
  v8f acc01 = {};  // tile0, cols 16..31
  v8f acc10 = {};  // tile1, cols 0..15
  v8f acc11 = {};  // tile1, cols 16..31
  float rsum0 = 0.f, rsum1 = 0.f;

  const float* Hp0 = H + (size_t)(v0 + i) * NE + ebeg + 2 * h;       // row v0+i
  const float* Hp1 = H + (size_t)(v0 + 16 + i) * NE + ebeg + 2 * h;  // row v0+16+i
  const float* yep = ye + (size_t)(ebeg + 2 * h) * COUT;

  for (int it = 0; it < CH2 / 4; ++it) {
    v2f a0 = *(const v2f*)Hp0;  // H[v0+i,    e+2h..+1], 8B aligned
    v2f a1 = *(const v2f*)Hp1;  // H[v0+16+i, e+2h..+1]
    __builtin_prefetch(Hp0 + 64, 0, 0);
    __builtin_prefetch(Hp1 + 64, 0, 0);
    rsum0 += a0.x + a0.y;       // fused row-degree partials
    rsum1 += a1.x + a1.y;
    v2f b0, b1;                 // shared B fragments (ye rows, L2-resident)
    b0.x = yep[i];              b0.y = yep[COUT + i];
    b1.x = yep[16 + i];         b1.y = yep[COUT + 16 + i];
    acc00 = wmma_f32(a0, b0, acc00);
    acc01 = wmma_f32(a0, b1, acc01);
    acc10 = wmma_f32(a1, b0, acc10);
    acc11 = wmma_f32(a1, b1, acc11);
    Hp0 += 4;
    Hp1 += 4;
    yep += 4 * COUT;
  }

  rsum0 += __shfl_xor(rsum0, 16, 32);
  rsum1 += __shfl_xor(rsum1, 16, 32);
  if (h == 0) {
    atomicAdd(rowsum + v0 + i,      rsum0);
    atomicAdd(rowsum + v0 + 16 + i, rsum1);
  }

#pragma unroll
  for (int r = 0; r < 8; ++r) {
    const int va = v0 + r + 8 * h;        // tile0 row
    const int vb = v0 + 16 + r + 8 * h;   // tile1 row
    atomicAdd(out + (size_t)va * COUT + i,      acc00[r]);
    atomicAdd(out + (size_t)va * COUT + 16 + i, acc01[r]);
    atomicAdd(out + (size_t)vb * COUT + i,      acc10[r]);
    atomicAdd(out + (size_t)vb * COUT + 16 + i, acc11[r]);
  }
}

// ---------------------------------------------------------------------------
// Scale: data[row, c] /= denom[row]  (applies De after pass 1, Dv after pass 2)
// ---------------------------------------------------------------------------
__global__ void k_scale(float* __restrict__ data,
                        const float* __restrict__ denom, int n) {
  const int idx = blockIdx.x * blockDim.x + threadIdx.x;
  if (idx < n) data[idx] = data[idx] / denom[idx >> 5];  // 32 channels per row
}

extern "C" void kernel_launch(void* const* d_in, const int* in_sizes, int n_in,
                              void* d_out, int out_size, void* d_ws, size_t ws_size,
                              hipStream_t stream) {
  const float* x     = (const float*)d_in[0];
  const float* H     = (const float*)d_in[1];
  const float* theta = (const float*)d_in[2];
  const float* bias  = (const float*)d_in[3];
  float* out = (float*)d_out;

  // Workspace layout (floats): y [N*32] | ye [E*32] | colsum [E] | rowsum [N]
  float* y      = (float*)d_ws;
  float* ye     = y + (size_t)NV * COUT;
  float* colsum = ye + (size_t)NE * COUT;
  float* rowsum = colsum + NE;

  // Zero the accumulators (colsum/rowsum contiguous); d_out doubles as the
  // pass-2 accumulator. hipMemsetAsync is graph-capture safe.
  hipMemsetAsync(ye, 0, (size_t)NE * COUT * sizeof(float), stream);
  hipMemsetAsync(colsum, 0, (size_t)(NE + NV) * sizeof(float), stream);
  hipMemsetAsync(out, 0, (size_t)NV * COUT * sizeof(float), stream);

  k_gemm_y<<<NV / 16, 32, 0, stream>>>(x, theta, bias, y);
  k_pass1<<<dim3(NE / 16, NV / CH1), 32, 0, stream>>>(H, y, ye, colsum);
  k_scale<<<(NE * COUT + 255) / 256, 256, 0, stream>>>(ye, colsum, NE * COUT);
  k_pass2<<<dim3(NV / 32, NE / CH2), 32, 0, stream>>>(H, ye, out, rowsum);
  k_scale<<<(NV * COUT + 255) / 256, 256, 0, stream>>>(out, rowsum, NV * COUT);
}